// QuantumConv2D_15315853377985
// MI455X (gfx1250) — compile-verified
//
#include <hip/hip_runtime.h>
#include <math.h>

#define DIM 16
#define PI_F 3.14159265358979323846f

typedef __attribute__((ext_vector_type(2))) float v2f;
typedef __attribute__((ext_vector_type(8))) float v8f;

// ---------------------------------------------------------------------------
// Kernel 0: build the 16x16 weight-circuit matrix M (column j = circuit * e_j).
// Qubit q lives at bit (3-q) of the state index (reference reshape order).
// ---------------------------------------------------------------------------
__global__ void qc_build_matrix(const float* __restrict__ qp, float* __restrict__ M) {
  const int j = threadIdx.x;
  if (j >= DIM) return;
  float st[DIM];
#pragma unroll
  for (int i = 0; i < DIM; ++i) st[i] = 0.0f;
  st[j] = 1.0f;

#pragma unroll
  for (int l = 0; l < 2; ++l) {
    // RY(weights[l][q]) on each qubit
#pragma unroll
    for (int q = 0; q < 4; ++q) {
      float h = 0.5f * qp[l * 4 + q];
      float s, c;
      sincosf(h, &s, &c);
      const int mask = 1 << (3 - q);
#pragma unroll
      for (int i = 0; i < DIM; ++i) {
        if (i & mask) continue;
        float s0 = st[i];
        float s1 = st[i | mask];
        st[i]        = c * s0 - s * s1;
        st[i | mask] = s * s0 + c * s1;
      }
    }
    // ring CNOTs: q -> (q+1)%4
#pragma unroll
    for (int q = 0; q < 4; ++q) {
      const int cm = 1 << (3 - q);
      const int tm = 1 << (3 - ((q + 1) & 3));
#pragma unroll
      for (int i = 0; i < DIM; ++i) {
        if ((i & cm) && !(i & tm)) {
          float t = st[i];
          st[i] = st[i | tm];
          st[i | tm] = t;
        }
      }
    }
  }
#pragma unroll
  for (int k = 0; k < DIM; ++k) M[k * DIM + j] = st[k];
}

// ---------------------------------------------------------------------------
// Main kernel: each wave processes 8 groups of 16 samples.
// For a group: build V (16 columns = product-state vectors), compute W = M*V
// with 4 chained V_WMMA_F32_16X16X4_F32, then out_n = sum_k z_k * W[k][n]^2.
// ---------------------------------------------------------------------------
__global__ __launch_bounds__(256)
void qc_wmma_main(const float* __restrict__ x, const float* __restrict__ M,
                  float* __restrict__ out, int nsamp) {
  const int lane = threadIdx.x & 31;
  const int wave = threadIdx.x >> 5;
  const int row  = lane & 15;   // A row / V column handled by this lane
  const int half = lane >> 4;   // lane half selects K offset (+2) and D row half (+8)

  // A-matrix chunks: a[t][p] = M[row][4t + 2*half + p]   (ISA 16x4 f32 layout)
  v2f a[4];
#pragma unroll
  for (int t = 0; t < 4; ++t) {
    const float* mp = M + row * 16 + 4 * t + 2 * half;
    a[t][0] = mp[0];
    a[t][1] = mp[1];
  }

  const int waveBase = blockIdx.x * 1024 + wave * 128;

  for (int g = 0; g < 8; ++g) {
    const int s  = waveBase + g * 16 + row;          // sample = column `row`
    const int ss = (s < nsamp) ? s : (nsamp - 1);    // clamp: keep EXEC full for WMMA

    // sample -> (b, oy, ox); patch = x[b, 0, 2oy:2oy+2, 2ox:2ox+2]
    const int ox = ss & 127;
    const int oy = (ss >> 7) & 127;
    const int b  = ss >> 14;
    const float* px = x + b * 65536 + oy * 512 + ox * 2;
    const float2 p01 = *reinterpret_cast<const float2*>(px);        // row 2oy
    const float2 p23 = *reinterpret_cast<const float2*>(px + 256);  // row 2oy+1

    // angles -> cos/sin of half-angles (qubit i <- patch element i)
    float pv[4] = {p01.x, p01.y, p23.x, p23.y};
    float cc[4], sn[4];
#pragma unroll
    for (int i = 0; i < 4; ++i) {
      float h = 0.5f * PI_F * tanhf(pv[i]);
      sn[i] = __sinf(h);
      cc[i] = __cosf(h);
    }

    // product vector v[r] = f(q0,bit3) f(q1,bit2) f(q2,bit1) f(q3,bit0)
    // pair products: ab[j] over bits(3,2) (qubits 0,1); cd[j] over bits(1,0) (qubits 2,3)
    float ab[4], cd[4];
#pragma unroll
    for (int j = 0; j < 4; ++j) {
      ab[j] = ((j & 2) ? sn[0] : cc[0]) * ((j & 1) ? sn[1] : cc[1]);
      cd[j] = ((j & 2) ? sn[2] : cc[2]) * ((j & 1) ? sn[3] : cc[3]);
    }
    // B chunk t, VGPR p holds V[4t + 2*half + p][column] : ab index = t, cd index = 2*half+p
    const float cdl0 = half ? cd[2] : cd[0];
    const float cdl1 = half ? cd[3] : cd[1];

    v8f acc = {0.0f, 0.0f, 0.0f, 0.0f, 0.0f, 0.0f, 0.0f, 0.0f};
#pragma unroll
    for (int t = 0; t < 4; ++t) {
      v2f bt;
      bt[0] = ab[t] * cdl0;
      bt[1] = ab[t] * cdl1;
      acc = __builtin_amdgcn_wmma_f32_16x16x4_f32(false, a[t], false, bt,
                                                  (short)0, acc, false, false);
    }

    // z-weighted reduction: D VGPR v holds row v (lanes<16) / v+8 (lanes>=16);
    // z = (-1)^(state index LSB) -> sign alternates with v in BOTH halves.
    float psum = 0.0f;
#pragma unroll
    for (int v = 0; v < 8; ++v) {
      float d = acc[v];
      psum += (v & 1) ? (-d * d) : (d * d);
    }
    psum += __shfl_xor(psum, 16, 32);   // combine row halves of this column

    if (half == 0 && s < nsamp) out[s] = psum;
  }
}

// ---------------------------------------------------------------------------
extern "C" void kernel_launch(void* const* d_in, const int* in_sizes, int n_in,
                              void* d_out, int out_size, void* d_ws, size_t ws_size,
                              hipStream_t stream) {
  const float* x  = (const float*)d_in[0];   // (B,1,256,256) f32
  const float* qp = (const float*)d_in[1];   // (2,4) f32
  float* out = (float*)d_out;                // (B,1,128,128) f32
  float* M   = (float*)d_ws;                 // 16x16 staging matrix

  const int B = in_sizes[0] / (256 * 256);
  const int nsamp = B * 128 * 128;

  qc_build_matrix<<<1, 32, 0, stream>>>(qp, M);

  const int samplesPerBlock = 1024;          // 8 waves * 8 groups * 16 samples
  const int nblocks = (nsamp + samplesPerBlock - 1) / samplesPerBlock;
  qc_wmma_main<<<nblocks, 256, 0, stream>>>(x, M, out, nsamp);
}